// DPOT2DLayer_49435073577224
// MI455X (gfx1250) — compile-verified
//
#include <hip/hip_runtime.h>
#include <math.h>

// DPOT2D spectral layer for gfx1250 (MI455X), all heavy math on
// V_WMMA_F32_16X16X4_F32 (fp32 matrix cores, wave32).
//
// Pipeline (B=2,H=W=256,C=512, kept=64 modes, 8 blocks x 64ch, hid=128):
//  1) k_fwd_w : pruned DFT along W with fused radix-2 DIF -> Y[h,kx,c]
//  2) k_fwd_h : pruned DFT along H with fused radix-2 DIF -> Z[b,ky,kx,c]
//  3) k_mlp1  : blockwise complex GEMM 64->128 + bias + exact GELU -> O1
//  4) k_mlp2  : blockwise complex GEMM 128->64 + bias -> O2[b,ky,kx,c]
//  5) k_inv_h : inverse DFT along H (64 modes -> 256) -> U[h,kx,c] (per-b)
//  6) k_inv_w : real inverse DFT along W (Hermitian weights) + residual -> out
//
// Radix-2 DIF (fused, no extra memory pass):
//   X[kx] = sum_{w<128} [x[w] + (-1)^kx x[w+128]] e^{-i 2pi kx w/256}
// Wave pairs own even/odd kx rows; B elements are built with one FMA from
// two loads, halving WMMA and LDS-lookup counts at equal HBM traffic.
//
// Twiddles: one 256-entry LDS table of pre-scaled (cos,sin) float2 pairs
// (every phase is a multiple of 2*pi/256) -> one ds_load_b64 per phase.
// Each wave computes a 16x32 output tile (2 N-tiles) so A-fragments are
// amortized over 2x the WMMAs.
//
// Workspace layout (floats), peak 134.2 MB:
//   [0        .. 4194304)  Zr      [4194304 .. 8388608)  Zi
//   [8388608  ..12582912)  O2r     [12582912..16777216)  O2i
//   [16777216 ..25165824)  R0a = Yr / O1r / Ur   (time-multiplexed)
//   [25165824 ..33554432)  R0b = Yi / O1i / Ui

typedef __attribute__((ext_vector_type(2))) float v2f;
typedef __attribute__((ext_vector_type(8))) float v8f;

__device__ __forceinline__ v8f wmma4(v2f a, v2f b, v8f c) {
  // D = A(16x4) * B(4x16) + C, fp32 matrix core.
  return __builtin_amdgcn_wmma_f32_16x16x4_f32(
      false, a, false, b, (short)0, c, false, false);
}

// tab[t] = (cos, sin)(2*pi*t/256) * scale
__device__ __forceinline__ void fill_cstab(float2* tab, float scale) {
  const float step = 6.283185307179586f / 256.0f;
  for (int t = threadIdx.x; t < 256; t += blockDim.x) {
    float s, c;
    sincosf(t * step, &s, &c);
    tab[t] = make_float2(c * scale, s * scale);
  }
  __syncthreads();
}

#define LANE (threadIdx.x & 31)
#define WAVE (threadIdx.x >> 5)

// ---- 1) forward DFT along W (radix-2 DIF fused):
// Y[h,kx,c] = (1/16) sum_{w<128} [x(w) + (-1)^kx x(w+128)] e^{-i*2pi*kx*w/256}
__global__ void k_fwd_w(const float* __restrict__ x, float* __restrict__ Yr,
                        float* __restrict__ Yi, int b) {
  __shared__ float2 tab[256];
  fill_cstab(tab, 0.0625f);
  const int lane = LANE;
  const int parity = WAVE >> 1;              // 0: even kx, 1: odd kx
  const int sub = WAVE & 1;                  // kx' tile within parity group
  const int m = 2 * (sub * 16 + (lane & 15)) + parity;   // actual kx
  const int n0 = blockIdx.x * 32 + (lane & 15);
  const int h  = blockIdx.y;                 // 256
  const int khalf = (lane >> 4) << 1;        // 0 | 2
  const float sg = parity ? -1.0f : 1.0f;
  const float* xrow = x + ((size_t)(b * 256 + h)) * (256 * 512);
  v8f aR0 = {}, aI0 = {}, aR1 = {}, aI1 = {};
  for (int k0 = 0; k0 < 128; k0 += 4) {
    const int ka = k0 + khalf, kb = ka + 1;
    const float2 cs0 = tab[(m * ka) & 255];
    const float2 cs1 = tab[(m * kb) & 255];
    v2f aC  = { cs0.x, cs1.x };
    v2f aSn = { -cs0.y, -cs1.y };
    const size_t pa = (size_t)ka * 512 + n0, pb = (size_t)kb * 512 + n0;
    const size_t q = (size_t)128 * 512;      // +128 rows of W
    v2f b0 = { fmaf(sg, xrow[pa + q], xrow[pa]),
               fmaf(sg, xrow[pb + q], xrow[pb]) };
    v2f b1 = { fmaf(sg, xrow[pa + q + 16], xrow[pa + 16]),
               fmaf(sg, xrow[pb + q + 16], xrow[pb + 16]) };
    aR0 = wmma4(aC,  b0, aR0);
    aI0 = wmma4(aSn, b0, aI0);
    aR1 = wmma4(aC,  b1, aR1);
    aI1 = wmma4(aSn, b1, aI1);
  }
  const int mbase = sub * 16 + ((lane >> 4) << 3);
  for (int r = 0; r < 8; ++r) {
    const int kx = 2 * (mbase + r) + parity;
    const size_t a = ((size_t)h * 64 + kx) * 512 + n0;
    Yr[a] = aR0[r];      Yi[a] = aI0[r];
    Yr[a + 16] = aR1[r]; Yi[a + 16] = aI1[r];
  }
}

// ---- 2) forward DFT along H (radix-2 DIF fused, complex input):
// Z[ky,kx,c] = (1/16) sum_{h<128} [Y(h) + (-1)^ky Y(h+128)] e^{-i*2pi*ky*h/256}
__global__ void k_fwd_h(const float* __restrict__ Yr, const float* __restrict__ Yi,
                        float* __restrict__ Zr, float* __restrict__ Zi, int b) {
  __shared__ float2 tab[256];
  fill_cstab(tab, 0.0625f);
  const int lane = LANE;
  const int parity = WAVE >> 1;              // 0: even ky, 1: odd ky
  const int sub = WAVE & 1;
  const int m = 2 * (sub * 16 + (lane & 15)) + parity;   // actual ky
  const int n0 = blockIdx.x * 32 + (lane & 15);
  const int kx = blockIdx.y;                 // 64
  const int khalf = (lane >> 4) << 1;
  const float sg = parity ? -1.0f : 1.0f;
  v8f aR0 = {}, aI0 = {}, aR1 = {}, aI1 = {};
  for (int k0 = 0; k0 < 128; k0 += 4) {
    const int ha = k0 + khalf, hb = ha + 1;
    const float2 cs0 = tab[(m * ha) & 255];
    const float2 cs1 = tab[(m * hb) & 255];
    v2f aC  = { cs0.x, cs1.x };
    v2f aS  = { cs0.y, cs1.y };
    v2f aSn = { -cs0.y, -cs1.y };
    const size_t pa = ((size_t)ha * 64 + kx) * 512 + n0;
    const size_t pb = ((size_t)hb * 64 + kx) * 512 + n0;
    const size_t q = (size_t)128 * 64 * 512;  // +128 rows of H
    v2f br0 = { fmaf(sg, Yr[pa + q], Yr[pa]), fmaf(sg, Yr[pb + q], Yr[pb]) };
    v2f bi0 = { fmaf(sg, Yi[pa + q], Yi[pa]), fmaf(sg, Yi[pb + q], Yi[pb]) };
    v2f br1 = { fmaf(sg, Yr[pa + q + 16], Yr[pa + 16]),
                fmaf(sg, Yr[pb + q + 16], Yr[pb + 16]) };
    v2f bi1 = { fmaf(sg, Yi[pa + q + 16], Yi[pa + 16]),
                fmaf(sg, Yi[pb + q + 16], Yi[pb + 16]) };
    // (c - i s)(Yr + i Yi): Re = c*Yr + s*Yi ; Im = c*Yi - s*Yr
    aR0 = wmma4(aC,  br0, aR0);  aR0 = wmma4(aS,  bi0, aR0);
    aI0 = wmma4(aC,  bi0, aI0);  aI0 = wmma4(aSn, br0, aI0);
    aR1 = wmma4(aC,  br1, aR1);  aR1 = wmma4(aS,  bi1, aR1);
    aI1 = wmma4(aC,  bi1, aI1);  aI1 = wmma4(aSn, br1, aI1);
  }
  const int mbase = sub * 16 + ((lane >> 4) << 3);
  for (int r = 0; r < 8; ++r) {
    const int ky = 2 * (mbase + r) + parity;
    const size_t a = (((size_t)(b * 64 + ky)) * 64 + kx) * 512 + n0;
    Zr[a] = aR0[r];      Zi[a] = aI0[r];
    Zr[a + 16] = aR1[r]; Zi[a + 16] = aI1[r];
  }
}

__device__ __forceinline__ float gelu_exact(float v) {
  return 0.5f * v * (1.0f + erff(v * 0.70710678118654752f));
}

// ---- 3) layer1: O1 = gelu(X @ (W1r + i W1i) + (b1r + i b1i)), per block nb
__global__ void k_mlp1(const float* __restrict__ Zr, const float* __restrict__ Zi,
                       const float* __restrict__ w1, const float* __restrict__ b1,
                       float* __restrict__ O1r, float* __restrict__ O1i) {
  const int lane = LANE;
  const int nb = blockIdx.z;                 // block 0..7
  const int mtile = blockIdx.y * 4 + WAVE;   // M=8192 -> 512 tiles
  const int p = mtile * 16 + (lane & 15);    // flat (b,ky,kx) position
  const int o0 = blockIdx.x * 32 + (lane & 15);
  const int khalf = (lane >> 4) << 1;
  v8f aR0 = {}, aI0 = {}, aR1 = {}, aI1 = {};
  const size_t xrow = (size_t)p * 512 + (size_t)nb * 64;
  const size_t wr0 = (size_t)nb * 64 * 128;        // w1[0][nb]
  const size_t wi0 = (size_t)(8 + nb) * 64 * 128;  // w1[1][nb]
  for (int k0 = 0; k0 < 64; k0 += 4) {
    const int ka = k0 + khalf, kb = ka + 1;
    v2f aXr = { Zr[xrow + ka], Zr[xrow + kb] };
    v2f aXi = { Zi[xrow + ka], Zi[xrow + kb] };
    v2f aXin = { -aXi.x, -aXi.y };
    const size_t ra = wr0 + (size_t)ka * 128 + o0, rb = wr0 + (size_t)kb * 128 + o0;
    const size_t ia = wi0 + (size_t)ka * 128 + o0, ib = wi0 + (size_t)kb * 128 + o0;
    v2f bWr0 = { w1[ra], w1[rb] },           bWi0 = { w1[ia], w1[ib] };
    v2f bWr1 = { w1[ra + 16], w1[rb + 16] }, bWi1 = { w1[ia + 16], w1[ib + 16] };
    aR0 = wmma4(aXr,  bWr0, aR0);  aR0 = wmma4(aXin, bWi0, aR0);
    aI0 = wmma4(aXi,  bWr0, aI0);  aI0 = wmma4(aXr,  bWi0, aI0);
    aR1 = wmma4(aXr,  bWr1, aR1);  aR1 = wmma4(aXin, bWi1, aR1);
    aI1 = wmma4(aXi,  bWr1, aI1);  aI1 = wmma4(aXr,  bWi1, aI1);
  }
  const float br0 = b1[(size_t)nb * 128 + o0];
  const float bi0 = b1[(size_t)(8 + nb) * 128 + o0];
  const float br1 = b1[(size_t)nb * 128 + o0 + 16];
  const float bi1 = b1[(size_t)(8 + nb) * 128 + o0 + 16];
  const int mbase = mtile * 16 + ((lane >> 4) << 3);
  for (int r = 0; r < 8; ++r) {
    const size_t a = ((size_t)nb * 8192 + (mbase + r)) * 128 + o0;
    O1r[a] = gelu_exact(aR0[r] + br0);
    O1i[a] = gelu_exact(aI0[r] + bi0);
    O1r[a + 16] = gelu_exact(aR1[r] + br1);
    O1i[a + 16] = gelu_exact(aI1[r] + bi1);
  }
}

// ---- 4) layer2: O2 = O1 @ (W2r + i W2i) + (b2r + i b2i), scatter to [p, C] planes
__global__ void k_mlp2(const float* __restrict__ O1r, const float* __restrict__ O1i,
                       const float* __restrict__ w2, const float* __restrict__ b2,
                       float* __restrict__ O2r, float* __restrict__ O2i) {
  const int lane = LANE;
  const int nb = blockIdx.z;
  const int mtile = blockIdx.y * 4 + WAVE;   // 512 tiles
  const int p = mtile * 16 + (lane & 15);
  const int o0 = blockIdx.x * 32 + (lane & 15);
  const int khalf = (lane >> 4) << 1;
  v8f aR0 = {}, aI0 = {}, aR1 = {}, aI1 = {};
  const size_t xrow = ((size_t)nb * 8192 + p) * 128;
  const size_t wr0 = (size_t)nb * 128 * 64;
  const size_t wi0 = (size_t)(8 + nb) * 128 * 64;
  for (int k0 = 0; k0 < 128; k0 += 4) {
    const int ka = k0 + khalf, kb = ka + 1;
    v2f aOr = { O1r[xrow + ka], O1r[xrow + kb] };
    v2f aOi = { O1i[xrow + ka], O1i[xrow + kb] };
    v2f aOin = { -aOi.x, -aOi.y };
    const size_t ra = wr0 + (size_t)ka * 64 + o0, rb = wr0 + (size_t)kb * 64 + o0;
    const size_t ia = wi0 + (size_t)ka * 64 + o0, ib = wi0 + (size_t)kb * 64 + o0;
    v2f bWr0 = { w2[ra], w2[rb] },           bWi0 = { w2[ia], w2[ib] };
    v2f bWr1 = { w2[ra + 16], w2[rb + 16] }, bWi1 = { w2[ia + 16], w2[ib + 16] };
    aR0 = wmma4(aOr,  bWr0, aR0);  aR0 = wmma4(aOin, bWi0, aR0);
    aI0 = wmma4(aOi,  bWr0, aI0);  aI0 = wmma4(aOr,  bWi0, aI0);
    aR1 = wmma4(aOr,  bWr1, aR1);  aR1 = wmma4(aOin, bWi1, aR1);
    aI1 = wmma4(aOi,  bWr1, aI1);  aI1 = wmma4(aOr,  bWi1, aI1);
  }
  const float br0 = b2[(size_t)nb * 64 + o0];
  const float bi0 = b2[(size_t)(8 + nb) * 64 + o0];
  const float br1 = b2[(size_t)nb * 64 + o0 + 16];
  const float bi1 = b2[(size_t)(8 + nb) * 64 + o0 + 16];
  const int mbase = mtile * 16 + ((lane >> 4) << 3);
  for (int r = 0; r < 8; ++r) {
    const size_t a = (size_t)(mbase + r) * 512 + (size_t)nb * 64 + o0;
    O2r[a] = aR0[r] + br0;      O2i[a] = aI0[r] + bi0;
    O2r[a + 16] = aR1[r] + br1; O2i[a + 16] = aI1[r] + bi1;
  }
}

// ---- 5) inverse DFT along H: U[h,kx,c] = (1/16) sum_{ky<64} e^{+i*2pi*ky*h/256} * O2
__global__ void k_inv_h(const float* __restrict__ O2r, const float* __restrict__ O2i,
                        float* __restrict__ Ur, float* __restrict__ Ui, int b) {
  __shared__ float2 tab[256];
  fill_cstab(tab, 0.0625f);
  const int lane = LANE;
  const int mtile = blockIdx.y * 4 + WAVE;   // h: 16 tiles
  const int m = mtile * 16 + (lane & 15);    // h
  const int n0 = blockIdx.x * 32 + (lane & 15);
  const int kx = blockIdx.z;                 // 64
  const int khalf = (lane >> 4) << 1;
  v8f aR0 = {}, aI0 = {}, aR1 = {}, aI1 = {};
  for (int k0 = 0; k0 < 64; k0 += 4) {
    const int ka = k0 + khalf, kb = ka + 1;
    const float2 cs0 = tab[(m * ka) & 255];
    const float2 cs1 = tab[(m * kb) & 255];
    v2f aC  = { cs0.x, cs1.x };
    v2f aS  = { cs0.y, cs1.y };
    v2f aSn = { -cs0.y, -cs1.y };
    const size_t pa = ((size_t)(b * 4096) + (size_t)ka * 64 + kx) * 512 + n0;
    const size_t pb = ((size_t)(b * 4096) + (size_t)kb * 64 + kx) * 512 + n0;
    v2f br0 = { O2r[pa], O2r[pb] },           bi0 = { O2i[pa], O2i[pb] };
    v2f br1 = { O2r[pa + 16], O2r[pb + 16] }, bi1 = { O2i[pa + 16], O2i[pb + 16] };
    // (c + i s)(Or + i Oi): Re = c*Or - s*Oi ; Im = c*Oi + s*Or
    aR0 = wmma4(aC,  br0, aR0);  aR0 = wmma4(aSn, bi0, aR0);
    aI0 = wmma4(aC,  bi0, aI0);  aI0 = wmma4(aS,  br0, aI0);
    aR1 = wmma4(aC,  br1, aR1);  aR1 = wmma4(aSn, bi1, aR1);
    aI1 = wmma4(aC,  bi1, aI1);  aI1 = wmma4(aS,  br1, aI1);
  }
  const int mbase = mtile * 16 + ((lane >> 4) << 3);
  for (int r = 0; r < 8; ++r) {
    const size_t a = ((size_t)(mbase + r) * 64 + kx) * 512 + n0;
    Ur[a] = aR0[r];      Ui[a] = aI0[r];
    Ur[a + 16] = aR1[r]; Ui[a + 16] = aI1[r];
  }
}

// ---- 6) real inverse DFT along W (+ residual):
// out[h,w,c] = x + (1/16)[ Ur[h,0,c] + sum_{kx=1}^{63} 2*(Ur cos - Ui sin) ]
// Table pre-scaled by 2/16; kx=0 over-count fixed by a scalar post-correction
// (w-independent since cos(0)=1, sin(0)=0).
__global__ void k_inv_w(const float* __restrict__ Ur, const float* __restrict__ Ui,
                        const float* __restrict__ x, float* __restrict__ out, int b) {
  __shared__ float2 tab[256];
  fill_cstab(tab, 0.125f);
  const int lane = LANE;
  const int mtile = blockIdx.y * 4 + WAVE;   // w: 16 tiles
  const int m = mtile * 16 + (lane & 15);    // w
  const int n0 = blockIdx.x * 32 + (lane & 15);
  const int h = blockIdx.z;                  // 256
  const int khalf = (lane >> 4) << 1;
  v8f acc0 = {}, acc1 = {};
  for (int k0 = 0; k0 < 64; k0 += 4) {
    const int ka = k0 + khalf, kb = ka + 1;
    const float2 cs0 = tab[(m * ka) & 255];
    const float2 cs1 = tab[(m * kb) & 255];
    v2f aC  = { cs0.x, cs1.x };
    v2f aSn = { -cs0.y, -cs1.y };
    const size_t pa = ((size_t)h * 64 + ka) * 512 + n0;
    const size_t pb = ((size_t)h * 64 + kb) * 512 + n0;
    v2f br0 = { Ur[pa], Ur[pb] },           bi0 = { Ui[pa], Ui[pb] };
    v2f br1 = { Ur[pa + 16], Ur[pb + 16] }, bi1 = { Ui[pa + 16], Ui[pb + 16] };
    acc0 = wmma4(aC, br0, acc0);  acc0 = wmma4(aSn, bi0, acc0);
    acc1 = wmma4(aC, br1, acc1);  acc1 = wmma4(aSn, bi1, acc1);
  }
  // kx=0 was weighted 2/16 inside the loop but belongs at 1/16.
  const float c0a = 0.0625f * Ur[(size_t)h * 64 * 512 + n0];
  const float c0b = 0.0625f * Ur[(size_t)h * 64 * 512 + n0 + 16];
  const int mbase = mtile * 16 + ((lane >> 4) << 3);
  for (int r = 0; r < 8; ++r) {
    const size_t a = (((size_t)(b * 256 + h)) * 256 + (mbase + r)) * 512 + n0;
    out[a] = x[a] + acc0[r] - c0a;
    out[a + 16] = x[a + 16] + acc1[r] - c0b;
  }
}

extern "C" void kernel_launch(void* const* d_in, const int* in_sizes, int n_in,
                              void* d_out, int out_size, void* d_ws, size_t ws_size,
                              hipStream_t stream) {
  (void)in_sizes; (void)n_in; (void)out_size; (void)ws_size;
  const float* x  = (const float*)d_in[0];
  const float* w1 = (const float*)d_in[1];
  const float* b1 = (const float*)d_in[2];
  const float* w2 = (const float*)d_in[3];
  const float* b2 = (const float*)d_in[4];
  float* out = (float*)d_out;

  float* f   = (float*)d_ws;               // needs >= 134,217,728 bytes
  float* Zr  = f;                          // 4,194,304 floats
  float* Zi  = f + 4194304;
  float* O2r = f + 8388608;
  float* O2i = f + 12582912;
  float* R0a = f + 16777216;               // Yr / O1r / Ur (time-multiplexed)
  float* R0b = f + 25165824;               // Yi / O1i / Ui

  const dim3 blk(128);                     // 4 wave32 per block
  for (int b = 0; b < 2; ++b) {
    k_fwd_w<<<dim3(16, 256), blk, 0, stream>>>(x, R0a, R0b, b);
    k_fwd_h<<<dim3(16, 64),  blk, 0, stream>>>(R0a, R0b, Zr, Zi, b);
  }
  k_mlp1<<<dim3(4, 128, 8), blk, 0, stream>>>(Zr, Zi, w1, b1, R0a, R0b);
  k_mlp2<<<dim3(2, 128, 8), blk, 0, stream>>>(R0a, R0b, w2, b2, O2r, O2i);
  for (int b = 0; b < 2; ++b) {
    k_inv_h<<<dim3(16, 4, 64),  blk, 0, stream>>>(O2r, O2i, R0a, R0b, b);
    k_inv_w<<<dim3(16, 4, 256), blk, 0, stream>>>(R0a, R0b, x, out, b);
  }
}